// HydraMultivariateGPU_27805618274845
// MI455X (gfx1250) — compile-verified
//
#include <hip/hip_runtime.h>

typedef __attribute__((ext_vector_type(2))) float v2f;
typedef __attribute__((ext_vector_type(4))) float v4f;
typedef __attribute__((ext_vector_type(8))) float v8f;

#define N_C   12
#define N_L   2048
#define N_D   8
#define N_DIV 2
#define N_H   32
#define N_K   8
#define N_CP  6
#define N_KW  9
#define PAD   512                       // max halo: 4 * 2^7 (largest dilation)
#define INP_LEN (N_L + 2 * PAD)         // 3072 floats
#define FEAT  (N_D * N_DIV * 2 * N_H * N_K)   // 8192 features per sample

// One block per (n, di, df, h). 256 threads = 8 waves (wave32).
__global__ __launch_bounds__(256) void hydra_wmma_kernel(
    const float* __restrict__ X,   // [N, C, L]
    const float* __restrict__ W,   // [D, DIV, K*H, 1, KW]
    const int*   __restrict__ I,   // [D, DIV, H, CP]
    float*       __restrict__ out) // [N, FEAT]
{
    __shared__ __align__(16) float S[N_L];        // raw gathered sum
    __shared__ __align__(16) float INP[INP_LEN];  // zero-padded conv input
    __shared__ __align__(16) float T[8][16][8];   // per-wave transpose tile [wave][t][k]
    __shared__ float accLds[2][N_K];              // [0]=cmax sums, [1]=cmin counts

    const int bid = blockIdx.x;
    const int h   =  bid        & (N_H   - 1);
    const int df  = (bid >> 5)  & (N_DIV - 1);
    const int di  = (bid >> 6)  & (N_D   - 1);
    const int n   =  bid >> 9;

    const int d    = 1 << di;          // dilation
    const int Lsrc = N_L - df;         // 2048 (raw) or 2047 (diff)
    const int Lout = Lsrc;             // "same" padding (p = 4d)

    const int tid = threadIdx.x;
    if (tid < 2 * N_K) accLds[tid >> 3][tid & 7] = 0.0f;

    // ---- Stage 1: zero halo, gather CP=6 channels (float4 loads), apply diff ----
    for (int t = tid; t < INP_LEN; t += 256) INP[t] = 0.0f;

    const int* Ig = I + ((di * N_DIV + df) * N_H + h) * N_CP;
    const int c0 = Ig[0], c1 = Ig[1], c2 = Ig[2], c3 = Ig[3], c4 = Ig[4], c5 = Ig[5];
    const float* Xn = X + (size_t)n * (N_C * N_L);
    const v4f* X0 = (const v4f*)(Xn + c0 * N_L);
    const v4f* X1 = (const v4f*)(Xn + c1 * N_L);
    const v4f* X2 = (const v4f*)(Xn + c2 * N_L);
    const v4f* X3 = (const v4f*)(Xn + c3 * N_L);
    const v4f* X4 = (const v4f*)(Xn + c4 * N_L);
    const v4f* X5 = (const v4f*)(Xn + c5 * N_L);
    v4f* Sv = (v4f*)S;
    for (int i = tid; i < N_L / 4; i += 256) {
        Sv[i] = X0[i] + X1[i] + X2[i] + X3[i] + X4[i] + X5[i];
    }
    __syncthreads();

    if (df) {
        for (int t = tid; t < N_L - 1; t += 256) INP[PAD + t] = S[t + 1] - S[t];
    } else {
        for (int t = tid; t < N_L; t += 256) INP[PAD + t] = S[t];
    }
    __syncthreads();

    // ---- Stage 2: grouped dilated conv via V_WMMA_F32_16X16X4_F32 ----
    const int lane   = tid & 31;
    const int wave   = tid >> 5;
    const int myK    = lane & 15;        // WMMA column (kernel id; valid < 8)
    const int halfHi = lane >> 4;        // A: taps {0,1} vs {2,3}; D: rows 0-7 vs 8-15
    const bool kValid = (myK < N_K);
    const int  tIdx   = lane & 15;       // time row this lane scans
    const bool doMax  = (halfHi == 0);   // lanes 0-15: max/argmax; 16-31: argmin count
    const float sgn   = doMax ? 1.0f : -1.0f;   // min(x) == max(-x), strict > keeps 1st idx

    // B operand (weights): 3 chunks of 4 taps; taps 9..11 and columns 8..15 are zero.
    const float* Wg = W + ((size_t)(di * N_DIV + df) * (N_K * N_H) + (size_t)h * N_K) * N_KW;
    v2f Bc[3];
#pragma unroll
    for (int ch = 0; ch < 3; ++ch) {
        const int tap = ch * 4 + (halfHi << 1);
        Bc[ch].x = (kValid && (tap     < N_KW)) ? Wg[myK * N_KW + tap    ] : 0.0f;
        Bc[ch].y = (kValid && (tap + 1 < N_KW)) ? Wg[myK * N_KW + tap + 1] : 0.0f;
    }

    const int oA = ((halfHi << 1) - 4) * d;   // chunk-0 tap offset for this lane
    const int d4 = d << 2;
    float* Tw = &T[wave][0][0];

    const int numTiles = (Lout + 15) >> 4;    // 128 tiles of 16 time steps
    for (int tile = wave; tile < numTiles; tile += 8) {
        const int t0   = tile << 4;
        const int base = PAD + t0 + tIdx;     // my A row (time) in padded INP

        // Unconditional LDS loads — halo zero-padded; chunk-2 hi-half feeds zero B rows.
        v2f a0, a1, a2;
        a0.x = INP[base + oA];
        a0.y = INP[base + oA + d];
        a1.x = INP[base + oA + d4];
        a1.y = INP[base + oA + d4 + d];
        a2.x = INP[base + d4];             // tap 8 (hi-half lanes hit zero B row anyway)
        a2.y = 0.0f;                       // taps 9/11: zero B rows

        v8f acc = {};
        acc = __builtin_amdgcn_wmma_f32_16x16x4_f32(false, a0, false, Bc[0], (short)0, acc, false, false);
        acc = __builtin_amdgcn_wmma_f32_16x16x4_f32(false, a1, false, Bc[1], (short)0, acc, false, false);
        acc = __builtin_amdgcn_wmma_f32_16x16x4_f32(false, a2, false, Bc[2], (short)0, acc, false, false);

        // Transpose D into per-wave LDS tile T[t][k] (lane = kernel col, VGPR r = time row).
        if (kValid) {
#pragma unroll
            for (int r = 0; r < 8; ++r)
                Tw[(halfHi << 6) + (r << 3) + myK] = acc[r];
        }

        // Row scan: lanes 0-15 find max/argmax of row tIdx, lanes 16-31 argmin (negated).
        const v4f* Trow = (const v4f*)&Tw[tIdx << 3];
        const v4f lo = Trow[0];
        const v4f hi = Trow[1];
        const float vv[8] = { sgn * lo.x, sgn * lo.y, sgn * lo.z, sgn * lo.w,
                              sgn * hi.x, sgn * hi.y, sgn * hi.z, sgn * hi.w };
        float best = vv[0];
        int   bk   = 0;
#pragma unroll
        for (int j = 1; j < 8; ++j) {
            const bool win = vv[j] > best;     // strict: first index wins ties
            best = win ? vv[j] : best;
            bk   = win ? j     : bk;
        }

        const int t = t0 + tIdx;
        if (t < Lout) {
            atomicAdd(&accLds[halfHi][bk], doMax ? best : 1.0f);
        }
    }

    // ---- Stage 3: write this block's 16 features ----
    __syncthreads();
    if (tid < N_K) {
        // feature index = (((di*DIV+df)*2 + s)*H + h)*K + k ; s=0 cmax, s=1 cmin
        const int fbase = ((di * N_DIV + df) * 2 * N_H + h) * N_K;
        float* o = out + (size_t)n * FEAT;
        o[fbase + tid]             = accLds[0][tid];
        o[fbase + N_H * N_K + tid] = accLds[1][tid];
    }
}

extern "C" void kernel_launch(void* const* d_in, const int* in_sizes, int n_in,
                              void* d_out, int out_size, void* d_ws, size_t ws_size,
                              hipStream_t stream) {
    const float* X = (const float*)d_in[0];
    const float* W = (const float*)d_in[1];
    const int*   I = (const int*)d_in[2];
    float* out = (float*)d_out;

    const int n = in_sizes[0] / (N_C * N_L);           // N = 128
    dim3 grid((unsigned)(n * N_D * N_DIV * N_H));      // 65536 blocks
    dim3 block(256);
    hipLaunchKernelGGL(hydra_wmma_kernel, grid, block, 0, stream, X, W, I, out);
}